// ICPEstimator_56770877718880
// MI455X (gfx1250) — compile-verified
//
#include <hip/hip_runtime.h>
#include <math.h>

#define IMG_H 720
#define IMG_W 1280
#define NPTS (IMG_H * IMG_W)
#define N_ITER 10
#define DIST_THR2 (13.3333333f * 13.3333333f)
#define NORMAL_THR 0.94f

typedef __attribute__((ext_vector_type(2))) float v2f;
typedef __attribute__((ext_vector_type(8))) float v8f;

// ---------------------------------------------------------------------------
// exp_se3: matches reference (Rodrigues + V matrix), small-angle branch
// ---------------------------------------------------------------------------
__device__ __forceinline__ void exp_se3_dev(const float x[6], float R[9], float t[3]) {
    float w0 = x[0], w1 = x[1], w2 = x[2];
    float v0 = x[3], v1 = x[4], v2 = x[5];
    float th2 = w0 * w0 + w1 * w1 + w2 * w2;
    bool  sml = th2 < 1e-10f;
    float th2s = sml ? 1.0f : th2;
    float th = sqrtf(th2s);
    float A = sml ? (1.0f - th2 * (1.0f / 6.0f))  : (sinf(th) / th);
    float B = sml ? (0.5f - th2 * (1.0f / 24.0f)) : ((1.0f - cosf(th)) / th2s);
    float C = sml ? ((1.0f / 6.0f) - th2 * (1.0f / 120.0f)) : ((1.0f - A) / th2s);
    float Wm[9] = { 0.f, -w2,  w1,
                    w2,  0.f, -w0,
                   -w1,  w0,  0.f };
    float W2[9];
    for (int i = 0; i < 3; ++i)
        for (int j = 0; j < 3; ++j) {
            float s = 0.f;
            for (int k = 0; k < 3; ++k) s += Wm[i * 3 + k] * Wm[k * 3 + j];
            W2[i * 3 + j] = s;
        }
    float V[9];
    for (int i = 0; i < 9; ++i) {
        float Iv = (i % 4 == 0) ? 1.f : 0.f;
        R[i] = Iv + A * Wm[i] + B * W2[i];
        V[i] = Iv + B * Wm[i] + C * W2[i];
    }
    t[0] = V[0] * v0 + V[1] * v1 + V[2] * v2;
    t[1] = V[3] * v0 + V[4] * v1 + V[5] * v2;
    t[2] = V[6] * v0 + V[7] * v1 + V[8] * v2;
}

// ---------------------------------------------------------------------------
// Init: zero accumulator and x, set R=I, t=tin=0
// ---------------------------------------------------------------------------
__global__ void icp_init_kernel(float* accum, float* x, float* Rt) {
    if (threadIdx.x == 0 && blockIdx.x == 0) {
        for (int i = 0; i < 64; ++i) accum[i] = 0.f;
        for (int i = 0; i < 6;  ++i) x[i] = 0.f;
        for (int i = 0; i < 9;  ++i) Rt[i] = (i % 4 == 0) ? 1.f : 0.f;
        for (int i = 9; i < 15; ++i) Rt[i] = 0.f;
    }
}

// ---------------------------------------------------------------------------
// Precompute ref_grid (backprojection) and ref_nrm (central diff with wrap,
// matching jnp.roll) directly from depth. One pass; 22 MB into workspace.
// ---------------------------------------------------------------------------
__global__ void icp_precompute_kernel(const float* __restrict__ depth,
                                      const float* __restrict__ Km,
                                      float* __restrict__ refGrid,
                                      float* __restrict__ refNrm) {
    int idx = blockIdx.x * blockDim.x + threadIdx.x;
    if (idx >= NPTS) return;
    int v = idx / IMG_W, u = idx - v * IMG_W;
    float fx = Km[0], cx = Km[2], fy = Km[4], cy = Km[5];
    float ifx = 1.f / fx, ify = 1.f / fy;

    auto bp = [&](int vv, int uu, float out[3]) {
        float d = depth[vv * IMG_W + uu];
        out[0] = ((float)uu - cx) * ifx * d;
        out[1] = ((float)vv - cy) * ify * d;
        out[2] = d;
    };

    float gp[3];
    bp(v, u, gp);
    refGrid[idx * 3 + 0] = gp[0];
    refGrid[idx * 3 + 1] = gp[1];
    refGrid[idx * 3 + 2] = gp[2];

    int um = (u == 0) ? (IMG_W - 1) : (u - 1);
    int up = (u == IMG_W - 1) ? 0 : (u + 1);
    int vm = (v == 0) ? (IMG_H - 1) : (v - 1);
    int vp = (v == IMG_H - 1) ? 0 : (v + 1);
    float pr[3], pl[3], pd[3], pu[3];
    bp(v, up, pr); bp(v, um, pl); bp(vp, u, pd); bp(vm, u, pu);
    float dx[3] = { (pr[0] - pl[0]) * 0.5f, (pr[1] - pl[1]) * 0.5f, (pr[2] - pl[2]) * 0.5f };
    float dy[3] = { (pd[0] - pu[0]) * 0.5f, (pd[1] - pu[1]) * 0.5f, (pd[2] - pu[2]) * 0.5f };
    float nx = dx[1] * dy[2] - dx[2] * dy[1];
    float ny = dx[2] * dy[0] - dx[0] * dy[2];
    float nz = dx[0] * dy[1] - dx[1] * dy[0];
    float inv = 1.f / (sqrtf(nx * nx + ny * ny + nz * nz) + 1e-12f);
    refNrm[idx * 3 + 0] = nx * inv;
    refNrm[idx * 3 + 1] = ny * inv;
    refNrm[idx * 3 + 2] = nz * inv;
}

// ---------------------------------------------------------------------------
// Main pass: per-point residual/Jacobian, Gram matrix G = sum a a^T via
// v_wmma_f32_16x16x4_f32. a = [J0..J5, r, w, 0...0] (16 padded).
// Each wave stages 32 a-vectors in LDS, then 8 rank-4 WMMA updates with
// identical A/B operand registers (columns of A == rows of B == a-vectors).
// ---------------------------------------------------------------------------
__global__ void __launch_bounds__(256)
icp_accum_kernel(const float* __restrict__ tp, const float* __restrict__ tn,
                 const unsigned char* __restrict__ mask,
                 const float* __restrict__ Km,
                 const float* __restrict__ refGrid, const float* __restrict__ refNrm,
                 const float* __restrict__ Rt,
                 float* accum, int kiter, int stride) {
    __shared__ __align__(16) float aLDS[8][32][16];  // 16 KB: 8 waves x 32 pts x 16
    __shared__ float redLDS[8][64];                  // 2 KB: per-wave 8x8 Gram corner

    const int wave = threadIdx.x >> 5;
    const int lane = threadIdx.x & 31;

    // Uniform transform state (R, t, tin = -(R^T t)) and intrinsics
    float R00 = Rt[0], R01 = Rt[1], R02 = Rt[2];
    float R10 = Rt[3], R11 = Rt[4], R12 = Rt[5];
    float R20 = Rt[6], R21 = Rt[7], R22 = Rt[8];
    float t0 = Rt[9], t1 = Rt[10], t2 = Rt[11];
    float ti0 = Rt[12], ti1 = Rt[13], ti2 = Rt[14];
    float fx = Km[0], cx = Km[2], fy = Km[4], cy = Km[5];

    v8f c = {0.f, 0.f, 0.f, 0.f, 0.f, 0.f, 0.f, 0.f};

    const int tid = blockIdx.x * blockDim.x + threadIdx.x;
    const int m = lane & 15;
    const int pairOff = (lane >> 4) * 2;

    // Zero-pad region (elements 8..15) never changes: store once, outside loop.
    float4* row = (float4*)aLDS[wave][lane];
    row[2] = make_float4(0.f, 0.f, 0.f, 0.f);
    row[3] = make_float4(0.f, 0.f, 0.f, 0.f);

    for (int k = 0; k < kiter; ++k) {
        int idx = tid + k * stride;
        int ci = (idx < NPTS) ? idx : (NPTS - 1);

        float px = tp[ci * 3 + 0], py = tp[ci * 3 + 1], pz = tp[ci * 3 + 2];
        float nx = tn[ci * 3 + 0], ny = tn[ci * 3 + 1], nz = tn[ci * 3 + 2];

        // camera frame: trg_c = R^T p + tin ; trg_n_c = R^T n
        float cxx = R00 * px + R10 * py + R20 * pz + ti0;
        float cyy = R01 * px + R11 * py + R21 * pz + ti1;
        float czz = R02 * px + R12 * py + R22 * pz + ti2;
        float nxc = R00 * nx + R10 * ny + R20 * nz;
        float nyc = R01 * nx + R11 * ny + R21 * nz;
        float nzc = R02 * nx + R12 * ny + R22 * nz;

        float zs = (czz > 1e-6f) ? czz : 1.0f;
        // One hardware v_rcp instead of two IEEE divide chains (zs ~ 100,
        // well-conditioned; only ULP-boundary pixels can flip validity).
        float invz = __builtin_amdgcn_rcpf(zs);
        float uu = fx * cxx * invz + cx;
        float vv = fy * cyy * invz + cy;
        bool valid = (czz > 1e-6f) && (vv < (float)IMG_H) && (uu < (float)IMG_W) &&
                     (vv > 0.f) && (uu > 0.f) && (idx < NPTS);

        int iu = (int)uu; iu = iu < 0 ? 0 : (iu > IMG_W - 1 ? IMG_W - 1 : iu);
        int iv = (int)vv; iv = iv < 0 ? 0 : (iv > IMG_H - 1 ? IMG_H - 1 : iv);
        int gi = iv * IMG_W + iu;
        valid = valid && (mask[gi] != 0);

        float spx = refGrid[gi * 3 + 0], spy = refGrid[gi * 3 + 1], spz = refGrid[gi * 3 + 2];
        float snx = refNrm[gi * 3 + 0],  sny = refNrm[gi * 3 + 1],  snz = refNrm[gi * 3 + 2];

        float dxx = spx - cxx, dyy = spy - cyy, dzz = spz - czz;
        valid = valid && ((dxx * dxx + dyy * dyy + dzz * dzz) < DIST_THR2);
        valid = valid && ((snx * nxc + sny * nyc + snz * nzc) > NORMAL_THR);
        float w = valid ? 1.f : 0.f;

        // world frame: p_w = R sp + t
        float pwx = R00 * spx + R01 * spy + R02 * spz + t0;
        float pwy = R10 * spx + R11 * spy + R12 * spz + t1;
        float pwz = R20 * spx + R21 * spy + R22 * spz + t2;
        float r = (nx * (pwx - px) + ny * (pwy - py) + nz * (pwz - pz)) * w;

        float a0 = (pwy * nz - pwz * ny) * w;  // cross(p_w, n)
        float a1 = (pwz * nx - pwx * nz) * w;
        float a2 = (pwx * ny - pwy * nx) * w;

        // Stage live part of the a-vector in LDS (pad rows already zeroed)
        row[0] = make_float4(a0, a1, a2, nx * w);
        row[1] = make_float4(ny * w, nz * w, r, w);

        // 8 rank-4 Gram updates: A(16x4) columns and B(4x16) rows are the same
        // a-vectors, so the operand register images coincide.
#pragma unroll
        for (int g = 0; g < 8; ++g) {
            v2f av;
            av.x = aLDS[wave][4 * g + pairOff + 0][m];
            av.y = aLDS[wave][4 * g + pairOff + 1][m];
            c = __builtin_amdgcn_wmma_f32_16x16x4_f32(false, av, false, av,
                                                      (short)0, c, false, false);
        }
    }

    // C/D layout: vgpr v, lanes 0-15 -> G[M=v][N=lane]. Keep 8x8 corner.
    if (lane < 8) {
#pragma unroll
        for (int v = 0; v < 8; ++v) redLDS[wave][v * 8 + lane] = c[v];
    }
    __syncthreads();
    if (threadIdx.x < 64) {
        float s = 0.f;
#pragma unroll
        for (int w2 = 0; w2 < 8; ++w2) s += redLDS[w2][threadIdx.x];
        atomicAdd(&accum[threadIdx.x], s);
    }
}

// ---------------------------------------------------------------------------
// Solve: JtJ dx = -Jtr (6x6 GE w/ partial pivot, LM damping), update x,
// refresh R/t/tin, re-zero accumulator for next pass.
// ---------------------------------------------------------------------------
__global__ void icp_solve_kernel(float* accum, float* x, float* Rt) {
    if (threadIdx.x != 0 || blockIdx.x != 0) return;
    float G[64];
    for (int i = 0; i < 64; ++i) { G[i] = accum[i]; accum[i] = 0.f; }

    float tr = 0.f;
    for (int i = 0; i < 6; ++i) tr += G[i * 8 + i];
    float lam = 1e-6f * tr;

    float A[6][7];
    for (int i = 0; i < 6; ++i) {
        for (int j = 0; j < 6; ++j) A[i][j] = G[i * 8 + j] + ((i == j) ? lam : 0.f);
        A[i][6] = -G[i * 8 + 6];  // -Jtr
    }
    for (int col = 0; col < 6; ++col) {
        int piv = col; float mx = fabsf(A[col][col]);
        for (int rr = col + 1; rr < 6; ++rr)
            if (fabsf(A[rr][col]) > mx) { mx = fabsf(A[rr][col]); piv = rr; }
        if (piv != col)
            for (int j = col; j < 7; ++j) { float tmp = A[col][j]; A[col][j] = A[piv][j]; A[piv][j] = tmp; }
        float d = A[col][col];
        float inv = (d != 0.f) ? 1.f / d : 0.f;
        for (int j = col; j < 7; ++j) A[col][j] *= inv;
        for (int rr = 0; rr < 6; ++rr) if (rr != col) {
            float f = A[rr][col];
            for (int j = col; j < 7; ++j) A[rr][j] -= f * A[col][j];
        }
    }
    float xn[6];
    for (int i = 0; i < 6; ++i) { xn[i] = x[i] + A[i][6]; x[i] = xn[i]; }

    float R[9], t[3];
    exp_se3_dev(xn, R, t);
    float ti0 = -(R[0] * t[0] + R[3] * t[1] + R[6] * t[2]);
    float ti1 = -(R[1] * t[0] + R[4] * t[1] + R[7] * t[2]);
    float ti2 = -(R[2] * t[0] + R[5] * t[1] + R[8] * t[2]);
    for (int i = 0; i < 9; ++i) Rt[i] = R[i];
    Rt[9] = t[0]; Rt[10] = t[1]; Rt[11] = t[2];
    Rt[12] = ti0; Rt[13] = ti1; Rt[14] = ti2;
}

// ---------------------------------------------------------------------------
// Final: T (4x4 row-major) and cost = sum(r^2)/max(sum(w),1)
// ---------------------------------------------------------------------------
__global__ void icp_final_kernel(const float* accum, const float* x, float* out) {
    if (threadIdx.x != 0 || blockIdx.x != 0) return;
    float xl[6];
    for (int i = 0; i < 6; ++i) xl[i] = x[i];
    float R[9], t[3];
    exp_se3_dev(xl, R, t);
    out[0] = R[0]; out[1] = R[1]; out[2]  = R[2]; out[3]  = t[0];
    out[4] = R[3]; out[5] = R[4]; out[6]  = R[5]; out[7]  = t[1];
    out[8] = R[6]; out[9] = R[7]; out[10] = R[8]; out[11] = t[2];
    out[12] = 0.f; out[13] = 0.f; out[14] = 0.f; out[15] = 1.f;
    float sr2 = accum[6 * 8 + 6];  // G[6][6] = sum r^2
    float sw  = accum[7 * 8 + 7];  // G[7][7] = sum w
    out[16] = sr2 / fmaxf(sw, 1.0f);
}

// ---------------------------------------------------------------------------
extern "C" void kernel_launch(void* const* d_in, const int* in_sizes, int n_in,
                              void* d_out, int out_size, void* d_ws, size_t ws_size,
                              hipStream_t stream) {
    const float*         d_depth = (const float*)d_in[0];
    const float*         d_tp    = (const float*)d_in[1];
    const float*         d_tn    = (const float*)d_in[2];
    const unsigned char* d_mask  = (const unsigned char*)d_in[3];
    const float*         d_K     = (const float*)d_in[4];

    float* ws      = (float*)d_ws;
    float* refGrid = ws;                  // 3*NPTS floats
    float* refNrm  = ws + 3 * NPTS;       // 3*NPTS floats
    float* accum   = ws + 6 * NPTS;       // 64 floats (8x8 Gram corner)
    float* xv      = accum + 64;          // 6 floats
    float* Rt      = xv + 6;              // R(9), t(3), tin(3)

    icp_init_kernel<<<1, 1, 0, stream>>>(accum, xv, Rt);
    icp_precompute_kernel<<<(NPTS + 255) / 256, 256, 0, stream>>>(d_depth, d_K, refGrid, refNrm);

    const int tpb = 256;
    const int blocks = 1440;                       // grid-stride; uniform trip count
    const int stride = blocks * tpb;
    const int kiter = (NPTS + stride - 1) / stride;

    for (int it = 0; it < N_ITER; ++it) {
        icp_accum_kernel<<<blocks, tpb, 0, stream>>>(d_tp, d_tn, d_mask, d_K,
                                                     refGrid, refNrm, Rt, accum,
                                                     kiter, stride);
        icp_solve_kernel<<<1, 1, 0, stream>>>(accum, xv, Rt);
    }
    // Final residual pass with converged x for cost
    icp_accum_kernel<<<blocks, tpb, 0, stream>>>(d_tp, d_tn, d_mask, d_K,
                                                 refGrid, refNrm, Rt, accum,
                                                 kiter, stride);
    icp_final_kernel<<<1, 1, 0, stream>>>(accum, xv, (float*)d_out);
}